// ExpandHarmonics_18416819765652
// MI455X (gfx1250) — compile-verified
//
#include <hip/hip_runtime.h>
#include <hip/hip_bf16.h>
#include <stdint.h>

// ---------------------------------------------------------------------------
// ExpandHarmonics for MI455X (gfx1250).
// Memory-bound ragged expansion: gcd -> n_max -> prefix sum -> expand rows.
// Meta rows (128B) are staged global -> LDS with the CDNA5 async engine and
// replicated n_max times with async LDS -> global stores (ASYNCcnt path).
// ---------------------------------------------------------------------------

#define BLK 256
#define DMETA_C 32

#if defined(__gfx1250__) && \
    __has_builtin(__builtin_amdgcn_global_load_async_to_lds_b128) && \
    __has_builtin(__builtin_amdgcn_global_store_async_from_lds_b128)
#define USE_ASYNC_LDS 1
#else
#define USE_ASYNC_LDS 0
#endif

#if USE_ASYNC_LDS
// Builtin parameter types (from clang diagnostic): pointer to 16-byte int
// vector in global (AS1) / LDS (AS3) address spaces.
typedef int v4i_t __attribute__((vector_size(16)));
typedef __attribute__((address_space(1))) v4i_t* g_b128_p;
typedef __attribute__((address_space(3))) v4i_t* l_b128_p;

__device__ __forceinline__ void wait_async0() {
#if __has_builtin(__builtin_amdgcn_s_wait_asynccnt)
  __builtin_amdgcn_s_wait_asynccnt(0);
#else
  asm volatile("s_wait_asynccnt 0" ::: "memory");
#endif
}
#endif

// Binary GCD for strictly positive small ints (inputs are 1..50): no integer
// division (clang otherwise expands % into a v_rcp-based division loop).
__device__ __forceinline__ int gcd_pos(int a, int b) {
  int az = __builtin_ctz(a);
  int bz = __builtin_ctz(b);
  int s = az < bz ? az : bz;
  a >>= az;
  while (b) {
    b >>= __builtin_ctz(b);
    if (a > b) { int t = a; a = b; b = t; }
    b -= a;
  }
  return a << s;
}

// ---------------------------------------------------------------------------
// Phase C: expand. Meta tile staged via async global->LDS, replicated via
// async LDS->global stores. Intra-block scan overlaps the async loads.
// d_out layout (float): hkl[3M] | dHKL[M] | wavelength[M] | meta[32M]
// (Defined first in the TU so its async instructions show in the disasm.)
// ---------------------------------------------------------------------------
__global__ void eh_phaseC(const int* __restrict__ hkl,
                          const float* __restrict__ dHKL,
                          const float* __restrict__ wavelength,
                          const float* __restrict__ meta,
                          const int* __restrict__ gnm,     // (g<<16)|n_max
                          const int* __restrict__ boff,
                          int N, int M,
                          float* __restrict__ out) {
  __shared__ float tile[BLK * DMETA_C];   // 32 KB meta staging tile
  __shared__ int sc[BLK];

  int t = threadIdx.x;
  int i = blockIdx.x * BLK + t;

#if USE_ASYNC_LDS
  // Kick off async global->LDS copy of this lane's 128B meta row (8 x b128).
  if (i < N) {
    const char* gsrc = (const char*)(meta + (size_t)i * DMETA_C);
    char* lrow = (char*)&tile[t * DMETA_C];
#pragma unroll
    for (int c = 0; c < 8; ++c) {
      __builtin_amdgcn_global_load_async_to_lds_b128(
          (g_b128_p)(gsrc + 16 * c),
          (l_b128_p)(lrow + 16 * c),
          0, 0);
    }
  }
#endif

  // Intra-block exclusive scan of n_max (overlaps with async loads).
  int packed = (i < N) ? gnm[i] : 0;
  int nm = packed & 0xFFFF;
  int g  = packed >> 16;
  sc[t] = nm;
  __syncthreads();
#pragma unroll
  for (int d = 1; d < BLK; d <<= 1) {
    int u = (t >= d) ? sc[t - d] : 0;
    __syncthreads();
    sc[t] += u;
    __syncthreads();
  }
  int off = boff[blockIdx.x] + (sc[t] - nm);

  float* out_hkl = out;
  float* out_d   = out + (size_t)3 * M;
  float* out_w   = out + (size_t)4 * M;
  float* out_m   = out + (size_t)5 * M;

#if !USE_ASYNC_LDS
  float4 mrow[8];
  if (i < N) {
    const float4* msrc = (const float4*)(meta + (size_t)i * DMETA_C);
#pragma unroll
    for (int c = 0; c < 8; ++c) mrow[c] = msrc[c];
  }
#endif

  if (i < N && nm > 0) {
    int h = hkl[3 * i + 0];
    int k = hkl[3 * i + 1];
    int l = hkl[3 * i + 2];
    int h0 = h / g, k0 = k / g, l0 = l / g;  // exact: g divides h,k,l
    float gf = (float)g;
    float d0 = dHKL[i] * gf;
    float w0 = wavelength[i] * gf;

#if USE_ASYNC_LDS
    wait_async0();   // meta row resident in LDS before LDS->global stores
#endif

    for (int n = 1; n <= nm; ++n) {
      int r = off + n - 1;
      float nf = (float)n;
      out_hkl[3 * (size_t)r + 0] = (float)(h0 * n);
      out_hkl[3 * (size_t)r + 1] = (float)(k0 * n);
      out_hkl[3 * (size_t)r + 2] = (float)(l0 * n);
      out_d[r] = d0 / nf;
      out_w[r] = w0 / nf;
#if USE_ASYNC_LDS
      char* gdst = (char*)(out_m + (size_t)r * DMETA_C);
      char* lrow = (char*)&tile[t * DMETA_C];
#pragma unroll
      for (int c = 0; c < 8; ++c) {
        __builtin_amdgcn_global_store_async_from_lds_b128(
            (g_b128_p)(gdst + 16 * c),
            (l_b128_p)(lrow + 16 * c),
            0, 0);
      }
#else
      float4* mdst = (float4*)(out_m + (size_t)r * DMETA_C);
#pragma unroll
      for (int c = 0; c < 8; ++c) mdst[c] = mrow[c];
#endif
    }
  }
#if USE_ASYNC_LDS
  else {
    wait_async0();   // keep the wave-wide wait uniform for inactive rows
  }
#endif
}

// ---------------------------------------------------------------------------
// Phase A: per-row gcd + n_max (packed (g<<16)|nm) + per-block sums.
// ---------------------------------------------------------------------------
__global__ void eh_phaseA(const int* __restrict__ hkl,
                          const float* __restrict__ dHKL,
                          const float* __restrict__ dmin,
                          int N,
                          int* __restrict__ gnm,
                          int* __restrict__ bsum) {
  __shared__ int red[BLK];
  int t = threadIdx.x;
  int i = blockIdx.x * BLK + t;
  int nm = 0;
  if (i < N) {
    int h = hkl[3 * i + 0];
    int k = hkl[3 * i + 1];
    int l = hkl[3 * i + 2];
    int g = gcd_pos(gcd_pos(h, k), l);
    float d0 = dHKL[i] * (float)g;
    nm = (int)floorf(d0 / dmin[0]);
    gnm[i] = (g << 16) | (nm & 0xFFFF);
  }
  red[t] = nm;
  __syncthreads();
#pragma unroll
  for (int s = BLK / 2; s > 0; s >>= 1) {
    if (t < s) red[t] += red[t + s];
    __syncthreads();
  }
  if (t == 0) bsum[blockIdx.x] = red[0];
}

// ---------------------------------------------------------------------------
// Phase B: exclusive scan of block sums (single block, chunked with carry).
// ---------------------------------------------------------------------------
__global__ void eh_phaseB(const int* __restrict__ bsum,
                          int* __restrict__ boff,
                          int nb) {
  __shared__ int s[1024];
  __shared__ int carry;
  int t = threadIdx.x;
  if (t == 0) carry = 0;
  __syncthreads();
  for (int base = 0; base < nb; base += 1024) {
    int i = base + t;
    int v = (i < nb) ? bsum[i] : 0;
    s[t] = v;
    __syncthreads();
#pragma unroll
    for (int d = 1; d < 1024; d <<= 1) {
      int u = (t >= d) ? s[t - d] : 0;
      __syncthreads();
      s[t] += u;
      __syncthreads();
    }
    if (i < nb) boff[i] = carry + (s[t] - v);   // exclusive
    __syncthreads();
    if (t == 1023) carry += s[1023];            // chunk total
    __syncthreads();
  }
}

// ---------------------------------------------------------------------------
// Launch
// ---------------------------------------------------------------------------
extern "C" void kernel_launch(void* const* d_in, const int* in_sizes, int n_in,
                              void* d_out, int out_size, void* d_ws, size_t ws_size,
                              hipStream_t stream) {
  const int*   hkl  = (const int*)d_in[0];
  const float* dh   = (const float*)d_in[1];
  const float* wl   = (const float*)d_in[2];
  const float* meta = (const float*)d_in[3];
  const float* dmin = (const float*)d_in[4];

  int N  = in_sizes[0] / 3;           // rows
  int M  = out_size / 37;             // expanded rows: 3+1+1+32 floats each
  int nb = (N + BLK - 1) / BLK;

  int* gnm  = (int*)d_ws;             // [N] packed (g<<16)|n_max
  int* bsum = gnm + N;                // [nb]
  int* boff = bsum + nb;              // [nb]

  eh_phaseA<<<nb, BLK, 0, stream>>>(hkl, dh, dmin, N, gnm, bsum);
  eh_phaseB<<<1, 1024, 0, stream>>>(bsum, boff, nb);
  eh_phaseC<<<nb, BLK, 0, stream>>>(hkl, dh, wl, meta, gnm, boff, N, M,
                                    (float*)d_out);
}